// MultiScaleDeformable3DAttention_64355789963832
// MI455X (gfx1250) — compile-verified
//
#include <hip/hip_runtime.h>
#include <hip/hip_bf16.h>
#include <math.h>

typedef float v2f __attribute__((ext_vector_type(2)));
typedef float v4f __attribute__((ext_vector_type(4)));
typedef float v8f __attribute__((ext_vector_type(8)));

#define NSCALE 3
#define NSAMP  8
#define DQ     256
#define QN     10000
#define BN     2
#define BQ     (BN * QN)

// ---------------------------------------------------------------------------
// Pack per-scale offset (24x256) + weight (8x256) projection matrices into one
// combined [96][256] weight matrix so the projection is a single WMMA GEMM.
// ---------------------------------------------------------------------------
__global__ void pack_w_kernel(const float* __restrict__ w_off0, const float* __restrict__ w_wt0,
                              const float* __restrict__ w_off1, const float* __restrict__ w_wt1,
                              const float* __restrict__ w_off2, const float* __restrict__ w_wt2,
                              float* __restrict__ wcomb)
{
    int t = blockIdx.x * blockDim.x + threadIdx.x;
    const int total = NSCALE * 32 * DQ;
    for (int idx = t; idx < total; idx += blockDim.x * gridDim.x) {
        int k  = idx % DQ;
        int r  = idx / DQ;        // 0..95
        int sc = r / 32;
        int rr = r % 32;
        const float* woff = (sc == 0) ? w_off0 : (sc == 1) ? w_off1 : w_off2;
        const float* wwt  = (sc == 0) ? w_wt0  : (sc == 1) ? w_wt1  : w_wt2;
        wcomb[idx] = (rr < 24) ? woff[rr * DQ + k] : wwt[(rr - 24) * DQ + k];
    }
}

// ---------------------------------------------------------------------------
// fp32 WMMA GEMM, LDS-staged, register-prefetched, spill-free.
//   C[m, col0+n] = sum_k A[m,k] * Bt[n,k] + bias[n]
//   A:  [M x K] row-major (lda = K);  Bt: [N x K] row-major (out = act @ W.T)
// Block = 256 threads (8 waves) -> 32(M) x 256(N) tile.
//   waves 0-3: m-tile 0, n-groups 0-3;  waves 4-7: m-tile 1, n-groups 0-3.
// Each wave: 4 x V_WMMA_F32_16X16X4_F32 accumulators (32 VGPRs) + <=20 VGPRs
// of staging -> no scratch spills (fixes round-3 Folded Spill/Reload).
// K in chunks of 16: coalesced b128 global prefetch of chunk c+1 overlaps the
// chunk-c WMMAs; fragments come from padded, conflict-free LDS rows.
// Fragment layout per CDNA5 ISA:
//   A: lane l holds A[mb + l%16][k + 2*(l/16) + {0,1}]
//   B: lane l holds B[k + 2*(l/16) + {0,1}][nb + l%16]
//   C: vgpr v, lane l -> row mb + v + 8*(l/16), col nb + l%16
// No thread ever early-returns (barriers + EXEC=all-ones for WMMA).
// ---------------------------------------------------------------------------
#define KC   16
#define LDSP (KC + 4)           // padded LDS row stride (80 B, 16 B aligned)

__global__ void gemm_wmma_lds(const float* __restrict__ A, int lda,
                              const float* __restrict__ Bt, int ldb,
                              const float* __restrict__ bias,
                              float* __restrict__ Cm, int ldc, int col0,
                              int M, int N, int K)
{
    __shared__ __align__(16) float Alds[32][LDSP];
    __shared__ __align__(16) float Blds[256][LDSP];

    const int tid  = threadIdx.x;     // 256
    const int lane = tid & 31;
    const int wave = tid >> 5;        // 0..7
    const int half = lane >> 4;       // 0: lanes 0-15, 1: lanes 16-31
    const int l16  = lane & 15;
    const int mi   = wave >> 2;       // m-tile within block (0 or 1)
    const int ng   = wave & 3;        // n-group (64 cols each)

    const int mbase = blockIdx.y * 32;
    const int nbase = blockIdx.x * 256;

    // --- global load assignments ---
    // A: 32 rows x 16 floats = 128 b128 -> threads 0..127 (wave-uniform split)
    const bool doA = (tid < 128);
    const int  ra  = (tid >> 2) & 31; // A row within tile
    const int  qa  = (tid & 3) * 4;   // quad column within chunk
    const float* __restrict__ Arow = A + (size_t)(mbase + ra) * lda;

    // B: 256 rows x 16 floats = 1024 b128 -> 4 per thread
    int rbrow[4];
    const float* __restrict__ Brow[4];
#pragma unroll
    for (int i = 0; i < 4; ++i) {
        const int rb = (tid >> 2) + i * 64;   // 0..255 within tile
        int n = nbase + rb;
        if (n > N - 1) n = N - 1;             // clamp loads in bounds
        rbrow[i] = rb;
        Brow[i]  = Bt + (size_t)n * ldb;
    }

    // --- preload chunk 0 into registers ---
    v4f areg = {};
    if (doA) areg = *(const v4f*)(Arow + qa);
    v4f breg[4];
#pragma unroll
    for (int i = 0; i < 4; ++i) breg[i] = *(const v4f*)(Brow[i] + qa);

    v8f acc[4] = {};
    const int nchunks = K / KC;
    for (int c = 0; c < nchunks; ++c) {
        // stage current chunk registers -> LDS
        if (doA) *(v4f*)&Alds[ra][qa] = areg;
#pragma unroll
        for (int i = 0; i < 4; ++i) *(v4f*)&Blds[rbrow[i]][qa] = breg[i];
        __syncthreads();

        // prefetch next chunk (global latency hidden behind the WMMAs)
        if (c + 1 < nchunks) {
            const int k0 = (c + 1) * KC;
            if (doA) areg = *(const v4f*)(Arow + k0 + qa);
#pragma unroll
            for (int i = 0; i < 4; ++i) breg[i] = *(const v4f*)(Brow[i] + k0 + qa);
        }

        // compute on staged chunk
#pragma unroll
        for (int ks = 0; ks < KC; ks += 4) {
            const int ko = ks + 2 * half;
            const v2f a = *(const v2f*)&Alds[mi * 16 + l16][ko];
#pragma unroll
            for (int t = 0; t < 4; ++t) {
                const v2f b = *(const v2f*)&Blds[ng * 64 + t * 16 + l16][ko];
                acc[t] = __builtin_amdgcn_wmma_f32_16x16x4_f32(
                    false, a, false, b, (short)0, acc[t], false, false);
            }
        }
        __syncthreads();
    }

    // --- epilogue: bias + store (guards are wave-uniform) ---
#pragma unroll
    for (int t = 0; t < 4; ++t) {
        const int ntile = ng * 64 + t * 16;
        if (nbase + ntile >= N) continue;     // wave-uniform
        const int ncol = nbase + ntile + l16;
        const float bv = bias ? bias[ncol] : 0.0f;
#pragma unroll
        for (int v = 0; v < 8; ++v) {
            const int row = mbase + mi * 16 + v + 8 * half;
            Cm[(size_t)row * ldc + col0 + ncol] = acc[t][v] + bv;
        }
    }
}

// ---------------------------------------------------------------------------
// Epilogue of the projection GEMM: tanh-scaled offsets -> clamped sample
// positions in [0,1], and softmax over the 8 per-scale attention weights.
// ---------------------------------------------------------------------------
__global__ void proj_epilogue(const float* __restrict__ proj,   // [BQ][96]
                              const float* __restrict__ refp,   // [BQ][3]
                              float* __restrict__ pos,          // [NSCALE][BQ][S][3]
                              float* __restrict__ wts)          // [NSCALE][BQ][S]
{
    const int bq = blockIdx.x * blockDim.x + threadIdx.x;
    if (bq >= BQ) return;
    const float r0 = refp[bq * 3 + 0];
    const float r1 = refp[bq * 3 + 1];
    const float r2 = refp[bq * 3 + 2];
    const float* p = proj + (size_t)bq * (NSCALE * 32);

#pragma unroll
    for (int sc = 0; sc < NSCALE; ++sc) {
        const float* ps = p + sc * 32;
#pragma unroll
        for (int s = 0; s < NSAMP; ++s) {
            float ox = tanhf(ps[s * 3 + 0]) * 0.25f;
            float oy = tanhf(ps[s * 3 + 1]) * 0.25f;
            float oz = tanhf(ps[s * 3 + 2]) * 0.25f;
            float px = fminf(fmaxf(r0 + ox, 0.0f), 1.0f);
            float py = fminf(fmaxf(r1 + oy, 0.0f), 1.0f);
            float pz = fminf(fmaxf(r2 + oz, 0.0f), 1.0f);
            float* dp = pos + (((size_t)sc * BQ + bq) * NSAMP + s) * 3;
            dp[0] = px; dp[1] = py; dp[2] = pz;
        }
        float mx = -3.402823e38f;
#pragma unroll
        for (int s = 0; s < NSAMP; ++s) mx = fmaxf(mx, ps[24 + s]);
        float e[NSAMP];
        float sum = 0.0f;
#pragma unroll
        for (int s = 0; s < NSAMP; ++s) { e[s] = __expf(ps[24 + s] - mx); sum += e[s]; }
        const float inv = 1.0f / sum;
#pragma unroll
        for (int s = 0; s < NSAMP; ++s)
            wts[((size_t)sc * BQ + bq) * NSAMP + s] = e[s] * inv;
    }
}

// ---------------------------------------------------------------------------
// LDS-tiled channel-first -> channel-last volume transpose:
//   volT[b][s][c] = vol[b][c][s],  s = linear (d*H + h)*W + w index.
// ---------------------------------------------------------------------------
#define TTC 32
#define TTS 64
__global__ void transpose_vol_kernel(const float* __restrict__ vin,
                                     float* __restrict__ vout,
                                     int C, int DHW)
{
    __shared__ float tile[TTC][TTS + 1];
    const int b  = blockIdx.z;
    const int c0 = blockIdx.y * TTC;
    const int s0 = blockIdx.x * TTS;
    const int t  = threadIdx.x;          // 256 threads
    const float* __restrict__ inp  = vin  + (size_t)b * C * DHW;
    float* __restrict__       outp = vout + (size_t)b * DHW * C;

#pragma unroll
    for (int i = 0; i < (TTC * TTS) / 256; ++i) {   // 8 iters
        int r  = t + i * 256;
        int cc = r / TTS, ss = r % TTS;             // ss fastest -> coalesced read
        tile[cc][ss] = inp[(size_t)(c0 + cc) * DHW + s0 + ss];
    }
    __syncthreads();
#pragma unroll
    for (int i = 0; i < (TTC * TTS) / 256; ++i) {
        int r  = t + i * 256;
        int ss = r / TTC, cc = r % TTC;             // cc fastest -> coalesced write
        outp[(size_t)(s0 + ss) * C + c0 + cc] = tile[cc][ss];
    }
}

// ---------------------------------------------------------------------------
// Trilinear sample + weighted aggregation, channel-LAST volume:
// lanes own consecutive channels -> each corner fetch by a wave is 1-4
// contiguous cachelines (vs 32 scattered lines in channel-first layout).
// ---------------------------------------------------------------------------
__global__ void sample_agg_cl_kernel(const float* __restrict__ volT,  // [B][DHW][C]
                                     const float* __restrict__ pos,   // [BQ][S][3]
                                     const float* __restrict__ wts,   // [BQ][S]
                                     float* __restrict__ agg,         // [BQ][C]
                                     int C, int D, int H, int W)
{
    __shared__ float red[256];
    const int bq  = blockIdx.x;
    const int b   = bq / QN;
    const int tid = threadIdx.x;
    const int c   = tid % C;
    const int g   = tid / C;
    const int groups = blockDim.x / C;

    const size_t DHW = (size_t)D * H * W;
    const float* __restrict__ vbase = volT + (size_t)b * DHW * C + c;

    float acc = 0.0f;
    for (int s = g; s < NSAMP; s += groups) {
        const float* pp = pos + ((size_t)bq * NSAMP + s) * 3;
        const float wgt = wts[(size_t)bq * NSAMP + s];
        const float ix = pp[0] * (float)W - 0.5f;
        const float iy = pp[1] * (float)H - 0.5f;
        const float iz = pp[2] * (float)D - 0.5f;
        const float x0f = floorf(ix), y0f = floorf(iy), z0f = floorf(iz);
        const float fx = ix - x0f, fy = iy - y0f, fz = iz - z0f;
        const int x0 = (int)x0f, y0 = (int)y0f, z0 = (int)z0f;
#pragma unroll
        for (int dz = 0; dz < 2; ++dz) {
            const int   zi = z0 + dz;
            const float wz = dz ? fz : 1.0f - fz;
            if ((unsigned)zi >= (unsigned)D) continue;
#pragma unroll
            for (int dy = 0; dy < 2; ++dy) {
                const int   yi = y0 + dy;
                const float wy = dy ? fy : 1.0f - fy;
                if ((unsigned)yi >= (unsigned)H) continue;
#pragma unroll
                for (int dx = 0; dx < 2; ++dx) {
                    const int   xi = x0 + dx;
                    const float wx = dx ? fx : 1.0f - fx;
                    if ((unsigned)xi >= (unsigned)W) continue;
                    const size_t lin = ((size_t)zi * H + yi) * W + xi;
                    acc += wgt * (wz * wy * wx) * vbase[lin * C];
                }
            }
        }
    }
    red[tid] = acc;
    __syncthreads();
    if (g == 0) {
        float t = acc;
        for (int gg = 1; gg < groups; ++gg) t += red[gg * C + c];
        agg[(size_t)bq * C + c] = t;
    }
}

// ---------------------------------------------------------------------------
// Fallback: channel-first sampling (used when ws too small to transpose).
// ---------------------------------------------------------------------------
__global__ void sample_agg_kernel(const float* __restrict__ vol,   // [B][C][D][H][W]
                                  const float* __restrict__ pos,
                                  const float* __restrict__ wts,
                                  float* __restrict__ agg,
                                  int C, int D, int H, int W)
{
    __shared__ float red[256];
    const int bq  = blockIdx.x;
    const int b   = bq / QN;
    const int tid = threadIdx.x;
    const int c   = tid % C;
    const int g   = tid / C;
    const int groups = blockDim.x / C;

    const size_t chanStride = (size_t)D * H * W;
    const float* __restrict__ vbase = vol + ((size_t)b * C + c) * chanStride;

    float acc = 0.0f;
    for (int s = g; s < NSAMP; s += groups) {
        const float* pp = pos + ((size_t)bq * NSAMP + s) * 3;
        const float wgt = wts[(size_t)bq * NSAMP + s];
        const float ix = pp[0] * (float)W - 0.5f;
        const float iy = pp[1] * (float)H - 0.5f;
        const float iz = pp[2] * (float)D - 0.5f;
        const float x0f = floorf(ix), y0f = floorf(iy), z0f = floorf(iz);
        const float fx = ix - x0f, fy = iy - y0f, fz = iz - z0f;
        const int x0 = (int)x0f, y0 = (int)y0f, z0 = (int)z0f;
#pragma unroll
        for (int dz = 0; dz < 2; ++dz) {
            const int   zi = z0 + dz;
            const float wz = dz ? fz : 1.0f - fz;
            if ((unsigned)zi >= (unsigned)D) continue;
#pragma unroll
            for (int dy = 0; dy < 2; ++dy) {
                const int   yi = y0 + dy;
                const float wy = dy ? fy : 1.0f - fy;
                if ((unsigned)yi >= (unsigned)H) continue;
                const float* rowp = vbase + ((size_t)zi * H + yi) * W;
#pragma unroll
                for (int dx = 0; dx < 2; ++dx) {
                    const int   xi = x0 + dx;
                    const float wx = dx ? fx : 1.0f - fx;
                    if ((unsigned)xi >= (unsigned)W) continue;
                    acc += wgt * (wz * wy * wx) * rowp[xi];
                }
            }
        }
    }
    red[tid] = acc;
    __syncthreads();
    if (g == 0) {
        float t = acc;
        for (int gg = 1; gg < groups; ++gg) t += red[gg * C + c];
        agg[(size_t)bq * C + c] = t;
    }
}

// ---------------------------------------------------------------------------
// In-place LayerNorm over the last dim (768), one block per row.
// ---------------------------------------------------------------------------
__global__ void layernorm_inplace(float* __restrict__ x,
                                  const float* __restrict__ gw,
                                  const float* __restrict__ gb,
                                  int N)
{
    __shared__ float red[256];
    const int row = blockIdx.x;
    const int tid = threadIdx.x;
    float* xr = x + (size_t)row * N;

    float s = 0.0f;
    for (int i = tid; i < N; i += 256) s += xr[i];
    red[tid] = s; __syncthreads();
    for (int o = 128; o > 0; o >>= 1) { if (tid < o) red[tid] += red[tid + o]; __syncthreads(); }
    const float mu = red[0] / (float)N;
    __syncthreads();

    float v = 0.0f;
    for (int i = tid; i < N; i += 256) { float d = xr[i] - mu; v += d * d; }
    red[tid] = v; __syncthreads();
    for (int o = 128; o > 0; o >>= 1) { if (tid < o) red[tid] += red[tid + o]; __syncthreads(); }
    const float inv = rsqrtf(red[0] / (float)N + 1e-5f);

    for (int i = tid; i < N; i += 256)
        xr[i] = (xr[i] - mu) * inv * gw[i] + gb[i];
}

// ---------------------------------------------------------------------------
extern "C" void kernel_launch(void* const* d_in, const int* in_sizes, int n_in,
                              void* d_out, int out_size, void* d_ws, size_t ws_size,
                              hipStream_t stream)
{
    (void)in_sizes; (void)n_in; (void)out_size;

    const float* queries = (const float*)d_in[0];
    const float* refp    = (const float*)d_in[1];
    const float* vol[NSCALE];
    const float* w_off[NSCALE]; const float* w_wt[NSCALE];
    const float* w_val[NSCALE]; const float* b_val[NSCALE];
    for (int i = 0; i < NSCALE; ++i) {
        const int base = 2 + i * 7;
        vol[i]   = (const float*)d_in[base + 0];
        w_off[i] = (const float*)d_in[base + 1];
        // base+2 = b_off (zeros), base+4 = b_wt (zeros) — folded out
        w_wt[i]  = (const float*)d_in[base + 3];
        w_val[i] = (const float*)d_in[base + 5];
        b_val[i] = (const float*)d_in[base + 6];
    }
    const float* ln_g  = (const float*)d_in[23];
    const float* ln_b  = (const float*)d_in[24];
    const float* w_out = (const float*)d_in[25];
    const float* b_out = (const float*)d_in[26];
    float* out = (float*)d_out;

    const int Cs[NSCALE] = {64, 128, 128};
    const int Ds[NSCALE] = {16, 8, 8};
    const int Hs[NSCALE] = {128, 64, 32};
    const int Wd[NSCALE] = {128, 64, 32};

    // Workspace carve-out (floats)
    float* ws    = (float*)d_ws;
    float* wcomb = ws;  ws += (size_t)NSCALE * 32 * DQ;        //   98 KB
    float* proj  = ws;  ws += (size_t)BQ * (NSCALE * 32);      //  7.7 MB
    float* pos   = ws;  ws += (size_t)NSCALE * BQ * NSAMP * 3; //  5.8 MB
    float* wt    = ws;  ws += (size_t)NSCALE * BQ * NSAMP;     //  1.9 MB
    float* agg0  = ws;  ws += (size_t)BQ * 64;                 //  5.1 MB
    float* agg1  = ws;  ws += (size_t)BQ * 128;                // 10.2 MB
    float* agg2  = ws;  ws += (size_t)BQ * 128;                // 10.2 MB
    float* xbuf  = ws;  ws += (size_t)BQ * (NSCALE * DQ);      // 61.4 MB
    float* aggp[NSCALE] = {agg0, agg1, agg2};

    // Optional channel-last volume copies (+176 MB), gated on ws_size.
    float* volT[NSCALE];
    size_t volElems[NSCALE];
    size_t need = (size_t)(ws - (float*)d_ws);
    for (int i = 0; i < NSCALE; ++i) {
        volElems[i] = (size_t)BN * Cs[i] * Ds[i] * Hs[i] * Wd[i];
        need += volElems[i];
    }
    const bool useCL = (ws_size >= need * sizeof(float));   // fixed per run -> deterministic
    if (useCL) {
        for (int i = 0; i < NSCALE; ++i) { volT[i] = ws; ws += volElems[i]; }
    }

    // 1) pack combined projection weights [96][256]
    pack_w_kernel<<<dim3(24), dim3(256), 0, stream>>>(
        w_off[0], w_wt[0], w_off[1], w_wt[1], w_off[2], w_wt[2], wcomb);

    // 1b) channel-last transposes (one-shot streaming, L2/HBM friendly)
    if (useCL) {
        for (int i = 0; i < NSCALE; ++i) {
            const int DHW = Ds[i] * Hs[i] * Wd[i];
            dim3 grid(DHW / TTS, Cs[i] / TTC, BN);
            transpose_vol_kernel<<<grid, 256, 0, stream>>>(vol[i], volT[i], Cs[i], DHW);
        }
    }

    // 2) projection GEMM: proj[BQ,96] = queries[BQ,256] @ wcomb.T
    {
        dim3 grid(1, BQ / 32);
        gemm_wmma_lds<<<grid, 256, 0, stream>>>(queries, DQ, wcomb, DQ, nullptr,
                                                proj, NSCALE * 32, 0,
                                                BQ, NSCALE * 32, DQ);
    }

    // 3) tanh/clip offsets + softmax weights
    proj_epilogue<<<(BQ + 255) / 256, 256, 0, stream>>>(proj, refp, pos, wt);

    // 4) trilinear sampling + per-sample weighted aggregation per scale
    for (int i = 0; i < NSCALE; ++i) {
        const float* posI = pos + (size_t)i * BQ * NSAMP * 3;
        const float* wtI  = wt  + (size_t)i * BQ * NSAMP;
        if (useCL) {
            sample_agg_cl_kernel<<<dim3(BQ), dim3(256), 0, stream>>>(
                volT[i], posI, wtI, aggp[i], Cs[i], Ds[i], Hs[i], Wd[i]);
        } else {
            sample_agg_kernel<<<dim3(BQ), dim3(256), 0, stream>>>(
                vol[i], posI, wtI, aggp[i], Cs[i], Ds[i], Hs[i], Wd[i]);
        }
    }

    // 5) value projections into concat buffer: xbuf[:, i*256:(i+1)*256]
    for (int i = 0; i < NSCALE; ++i) {
        dim3 grid(1, BQ / 32);
        gemm_wmma_lds<<<grid, 256, 0, stream>>>(aggp[i], Cs[i], w_val[i], Cs[i], b_val[i],
                                                xbuf, NSCALE * DQ, i * DQ,
                                                BQ, DQ, Cs[i]);
    }

    // 6) LayerNorm over 768, in place
    layernorm_inplace<<<dim3(BQ), dim3(256), 0, stream>>>(xbuf, ln_g, ln_b, NSCALE * DQ);

    // 7) output GEMM: out[BQ,256] = xbuf[BQ,768] @ w_out.T + b_out
    {
        dim3 grid(1, BQ / 32);
        gemm_wmma_lds<<<grid, 256, 0, stream>>>(xbuf, NSCALE * DQ, w_out, NSCALE * DQ, b_out,
                                                out, DQ, 0,
                                                BQ, DQ, NSCALE * DQ);
    }
}